// StiefelHouseholder_57818849739331
// MI455X (gfx1250) — compile-verified
//
#include <hip/hip_runtime.h>

typedef __attribute__((ext_vector_type(2))) float v2f;
typedef __attribute__((ext_vector_type(8))) float v8f;

#define B_     16
#define N_     4096
#define R_     256
#define NB     16            // reflector block size (WY block)
#define NBLK   (R_ / NB)     // 16 blocks
#define KT     64            // Q columns per workgroup
#define CT_    (KT / 16)     // 4 column tiles of 16
#define NWAVES 8
#define TPB    256

// v[n][i] = param[n][i] if n > i ; 1 if n == i ; 0 if n < i
// Written so the load is ALWAYS consumed -> unconditional global_load + 2 cndmask + fma
// (avoids the divergent branch-around-load the ternary form produced).
__device__ __forceinline__ float vget(const float* __restrict__ pb, int n, int i) {
  float x = pb[(size_t)n * R_ + i];          // always in-bounds: n<N_, i<R_
  float m = (n > i) ? 1.0f : 0.0f;
  float e = (n == i) ? 1.0f : 0.0f;
  return fmaf(x, m, e);
}

// ---------------- Kernel 1: Q = eye(N, R) broadcast over batch ----------------
__global__ __launch_bounds__(TPB) void k_init_eye(float* __restrict__ Q) {
  size_t idx = (size_t)blockIdx.x * TPB + threadIdx.x;
  int k = (int)(idx % R_);
  int n = (int)((idx / R_) % N_);
  Q[idx] = (n == k) ? 1.0f : 0.0f;
}

// ---------------- Kernel 2: tau[b,i] = 2 / (1 + sum_{n>i} p[b,n,i]^2) ----------------
__global__ __launch_bounds__(TPB) void k_tau(const float* __restrict__ p,
                                             float* __restrict__ tau) {
  int wg = blockIdx.x;
  int b = wg / R_, i = wg % R_;
  const float* pb = p + (size_t)b * N_ * R_;
  float acc = 0.0f;
  for (int n = i + 1 + (int)threadIdx.x; n < N_; n += TPB) {
    float x = pb[(size_t)n * R_ + i];
    acc += x * x;
  }
  __shared__ float red[TPB];
  red[threadIdx.x] = acc;
  __syncthreads();
  for (int s = TPB / 2; s > 0; s >>= 1) {
    if ((int)threadIdx.x < s) red[threadIdx.x] += red[threadIdx.x + s];
    __syncthreads();
  }
  if (threadIdx.x == 0) tau[b * R_ + i] = 2.0f / (1.0f + red[0]);
}

// ---------------- Kernel 3: build T (16x16 upper-tri, LAPACK larft forward) ----------------
__global__ __launch_bounds__(TPB) void k_T(const float* __restrict__ p,
                                           const float* __restrict__ tau,
                                           float* __restrict__ Tall) {
  int wg = blockIdx.x;
  int b = wg / NBLK, blk = wg % NBLK;
  int i0 = blk * NB;
  const float* pb = p + (size_t)b * N_ * R_;
  __shared__ float S[NB][NB];   // Gram: S[j1][j2] = v_{j1}^T v_{j2}, j1 < j2
  __shared__ float Tl[NB][NB];
  int j1 = threadIdx.x >> 4, j2 = threadIdx.x & 15;
  if (threadIdx.x < NB * NB) { S[j1][j2] = 0.0f; Tl[j1][j2] = 0.0f; }
  __syncthreads();
  if (threadIdx.x < NB * NB && j1 < j2) {
    float acc = 0.0f;
    for (int n = i0 + j2; n < N_; ++n)           // first nonzero product at n = i0+j2
      acc += vget(pb, n, i0 + j1) * vget(pb, n, i0 + j2);
    S[j1][j2] = acc;
  }
  __syncthreads();
  if (threadIdx.x == 0) {
    for (int j = 0; j < NB; ++j) {
      float tj = tau[b * R_ + i0 + j];
      for (int r = 0; r < j; ++r) {
        float acc = 0.0f;
        for (int c = r; c < j; ++c) acc += Tl[r][c] * S[c][j];
        Tl[r][j] = -tj * acc;
      }
      Tl[j][j] = tj;
    }
  }
  __syncthreads();
  if (threadIdx.x < NB * NB)
    Tall[(size_t)wg * NB * NB + threadIdx.x] = Tl[j1][j2];
}

// ---------------- Kernel 4: blocked larfb, one workgroup per (batch, 64-col tile) ----------------
__global__ __launch_bounds__(TPB) void k_apply(const float* __restrict__ p,
                                               const float* __restrict__ Tall,
                                               float* __restrict__ Q) {
  int b = blockIdx.x / (R_ / KT);
  int colbase = (blockIdx.x % (R_ / KT)) * KT;
  const float* pb = p + (size_t)b * N_ * R_;
  float* Qb = Q + (size_t)b * N_ * R_;

  int lane = threadIdx.x & 31, wave = threadIdx.x >> 5;
  int l16 = lane & 15, hi = lane >> 4, hi2 = hi << 1;

  __shared__ float Xs[NB][KT];   // X = V^T Q(cols)
  __shared__ float Ys[NB][KT];   // Ys = -(T X)
  __shared__ float Ts[NB * NB];  // T staged in LDS

  for (int blk = NBLK - 1; blk >= 0; --blk) {
    int i0 = blk * NB;

    // zero X partial-sum buffer + stage this block's T into LDS
    {
      const float* Tg = Tall + (size_t)(b * NBLK + blk) * NB * NB;
      for (int t = threadIdx.x; t < NB * KT; t += TPB) (&Xs[0][0])[t] = 0.0f;
      if (threadIdx.x < NB * NB) Ts[threadIdx.x] = Tg[threadIdx.x];
    }
    __syncthreads();

    // ---- Phase A: X = V^T * Q[:, cols]   (M=16 reflectors, N=16 cols, K over rows n)
    {
      v8f accX[CT_] = {};
      int nlo = wave * (N_ / NWAVES);
      int nhi = nlo + (N_ / NWAVES);
      int n0s = (nlo > i0) ? nlo : i0;       // rows < i0 contribute zero (i0 % 4 == 0)
      #pragma unroll 2
      for (int n0 = n0s; n0 < nhi; n0 += 4) {
        v2f a;                               // A[M=j][K=n] = v[n][i0+j]
        a.x = vget(pb, n0 + hi2,     i0 + l16);
        a.y = vget(pb, n0 + hi2 + 1, i0 + l16);
        #pragma unroll
        for (int ct = 0; ct < CT_; ++ct) {
          int col = colbase + ct * 16 + l16;
          v2f bb;                            // B[K=n][N=col] = Q[n][col]
          bb.x = Qb[(size_t)(n0 + hi2) * R_ + col];
          bb.y = Qb[(size_t)(n0 + hi2 + 1) * R_ + col];
          accX[ct] = __builtin_amdgcn_wmma_f32_16x16x4_f32(
              false, a, false, bb, (short)0, accX[ct], false, false);
        }
      }
      #pragma unroll
      for (int ct = 0; ct < CT_; ++ct)
        #pragma unroll
        for (int r = 0; r < 8; ++r)
          atomicAdd(&Xs[r + 8 * hi][ct * 16 + l16], accX[ct][r]);   // ds_add_f32
    }
    __syncthreads();

    // ---- Phase B: Ys = -(T @ Xs)   (16x16 * 16xKT, tiny, T from LDS)
    {
      for (int t = threadIdx.x; t < NB * KT; t += TPB) {
        int j = t >> 6, k = t & (KT - 1);    // KT == 64
        float acc = 0.0f;
        #pragma unroll
        for (int c = 0; c < NB; ++c) acc += Ts[j * NB + c] * Xs[c][k];
        Ys[j][k] = -acc;
      }
    }
    __syncthreads();

    // ---- Phase C: Q[:, cols] = Q[:, cols] + V * Ys   (WMMA with C = Q tile)
    {
      v2f bfr[CT_][4];                        // B[K=j][N=col] = Ys[j][col], hoisted
      #pragma unroll
      for (int ct = 0; ct < CT_; ++ct)
        #pragma unroll
        for (int ks = 0; ks < 4; ++ks) {
          bfr[ct][ks].x = Ys[ks * 4 + hi2][ct * 16 + l16];
          bfr[ct][ks].y = Ys[ks * 4 + hi2 + 1][ct * 16 + l16];
        }
      // first row-tile with any update is i0/16; align start to this wave's stride class
      int first = i0 >> 4;
      int d = first - wave;
      int mt0 = wave + ((d > 0) ? ((d + NWAVES - 1) / NWAVES) * NWAVES : 0);
      for (int mt = mt0; mt < N_ / 16; mt += NWAVES) {
        int n0 = mt * 16;
        v2f afr[4];                           // A[M=n][K=j] = v[n0+M][i0+j]
        #pragma unroll
        for (int ks = 0; ks < 4; ++ks) {
          afr[ks].x = vget(pb, n0 + l16, i0 + ks * 4 + hi2);
          afr[ks].y = vget(pb, n0 + l16, i0 + ks * 4 + hi2 + 1);
        }
        #pragma unroll
        for (int ct = 0; ct < CT_; ++ct) {
          int col = colbase + ct * 16 + l16;
          v8f c;
          #pragma unroll
          for (int r = 0; r < 8; ++r)
            c[r] = Qb[(size_t)(n0 + r + 8 * hi) * R_ + col];
          #pragma unroll
          for (int ks = 0; ks < 4; ++ks)
            c = __builtin_amdgcn_wmma_f32_16x16x4_f32(
                false, afr[ks], false, bfr[ct][ks], (short)0, c, false, false);
          #pragma unroll
          for (int r = 0; r < 8; ++r)
            Qb[(size_t)(n0 + r + 8 * hi) * R_ + col] = c[r];
        }
      }
    }
    __syncthreads();
  }
}

extern "C" void kernel_launch(void* const* d_in, const int* in_sizes, int n_in,
                              void* d_out, int out_size, void* d_ws, size_t ws_size,
                              hipStream_t stream) {
  const float* param = (const float*)d_in[0];
  float* Q = (float*)d_out;
  float* tau = (float*)d_ws;                  // B_*R_ floats
  float* T   = tau + B_ * R_;                 // B_*NBLK*NB*NB floats

  k_init_eye<<<dim3((B_ * N_ * R_) / TPB), dim3(TPB), 0, stream>>>(Q);
  k_tau    <<<dim3(B_ * R_),               dim3(TPB), 0, stream>>>(param, tau);
  k_T      <<<dim3(B_ * NBLK),             dim3(TPB), 0, stream>>>(param, tau, T);
  k_apply  <<<dim3(B_ * (R_ / KT)),        dim3(TPB), 0, stream>>>(param, T, Q);
}